// LatticeGnn_80041010528538
// MI455X (gfx1250) — compile-verified
//
#include <hip/hip_runtime.h>
#include <math.h>

// ---------------------------------------------------------------------------
// GCN link predictor for MI455X (gfx1250, wave32).
// Memory-bound edge scatter/gather via L2 float atomics; dense per-layer GEMMs
// via V_WMMA_F32_16X16X4_F32 (exact f32, matches reference numerics).
// ---------------------------------------------------------------------------

typedef __attribute__((ext_vector_type(2))) float v2f;
typedef __attribute__((ext_vector_type(8))) float v8f;

#define TPB 256

// ---- init / degree ---------------------------------------------------------

__global__ void k_fill(float* __restrict__ p, float v, int n) {
    int i = blockIdx.x * blockDim.x + threadIdx.x;
    if (i < n) p[i] = v;
}

__global__ void k_degree(const int* __restrict__ dst, float* __restrict__ deg, int E) {
    int e = blockIdx.x * blockDim.x + threadIdx.x;
    if (e < E) atomicAdd(&deg[dst[e]], 1.0f);
}

// deg >= 1 always (self-loop), so no zero guard needed (matches reference's
// where(deg>0) since deg>0 always holds here).
__global__ void k_rsqrt_inplace(float* __restrict__ deg, int n) {
    int i = blockIdx.x * blockDim.x + threadIdx.x;
    if (i < n) deg[i] = rsqrtf(deg[i]);
}

// ---- layer 1 GEMM: [N,1] @ [1,8] (trivial outer product) -------------------

__global__ void k_lin1(const float* __restrict__ x, const float* __restrict__ W1,
                       float* __restrict__ h, int n) {
    int i = blockIdx.x * blockDim.x + threadIdx.x;
    if (i >= n) return;
    float xv = x[i];
#pragma unroll
    for (int j = 0; j < 8; ++j) h[i * 8 + j] = xv * W1[j];
}

// ---- WMMA GEMM: h[N,OUTF] = in[N,K] @ W[K,OUTF], K%4==0, OUTF%16==0 --------
// One wave per 16-row tile. A layout (16x4 f32): lanes 0-15 -> M=lane, kh=0
// (K=0,1); lanes 16-31 -> M=lane-16, kh=2 (K=2,3). B (4x16) mirrors with
// N=lane&15. C/D (16x16 f32): 8 VGPRs; lane<16: N=lane, M=r; lane>=16: M=r+8.

template <int K, int OUTF>
__global__ void k_gemm_wmma(const float* __restrict__ in, const float* __restrict__ W,
                            float* __restrict__ out, int nrows) {
    int wave = (blockIdx.x * blockDim.x + threadIdx.x) >> 5;   // uniform per wave
    int lane = threadIdx.x & 31;
    int ntiles = (nrows + 15) >> 4;
    if (wave >= ntiles) return;                                // whole-wave exit
    int row0 = wave << 4;
    int m    = lane & 15;
    int kh   = (lane >> 4) << 1;                               // 0 or 2

    int arow = row0 + m;
    if (arow > nrows - 1) arow = nrows - 1;                    // clamp (EXEC stays full)

#pragma unroll
    for (int n0 = 0; n0 < OUTF; n0 += 16) {
        v8f c = {0.f, 0.f, 0.f, 0.f, 0.f, 0.f, 0.f, 0.f};
        int ncol = n0 + m;
#pragma unroll
        for (int kk = 0; kk < K; kk += 4) {
            v2f a, b;
            a.x = in[arow * K + kk + kh];
            a.y = in[arow * K + kk + kh + 1];
            b.x = W[(kk + kh) * OUTF + ncol];
            b.y = W[(kk + kh + 1) * OUTF + ncol];
            c = __builtin_amdgcn_wmma_f32_16x16x4_f32(
                    /*neg_a=*/false, a, /*neg_b=*/false, b,
                    /*c_mod=*/(short)0, c, /*reuse_a=*/false, /*reuse_b=*/false);
        }
        int mbase = (lane >> 4) << 3;
        if (row0 + 16 <= nrows) {
            // Fast path (always taken when nrows % 16 == 0): wave-uniform
            // branch, unguarded clause-batched stores, no exec-mask churn.
#pragma unroll
            for (int r = 0; r < 8; ++r)
                out[(row0 + mbase + r) * OUTF + ncol] = c[r];
        } else {
#pragma unroll
            for (int r = 0; r < 8; ++r) {
                int row = row0 + mbase + r;
                if (row < nrows) out[row * OUTF + ncol] = c[r];
            }
        }
    }
}

// ---- aggregation -----------------------------------------------------------

// agg[v*F+f] = h[v*F+f] * dinv[v]^2   (self-loop edge, also zero-initializes agg)
template <int LOGF>
__global__ void k_selfloop_init(const float* __restrict__ h, const float* __restrict__ dinv,
                                float* __restrict__ agg, int n) {
    constexpr int F = 1 << LOGF;
    int i = blockIdx.x * blockDim.x + threadIdx.x;
    if (i >= n * F) return;
    int v = i >> LOGF;
    float di = dinv[v];
    agg[i] = h[i] * di * di;
}

// F lanes per edge: agg[dst*F+f] += h[src*F+f] * dinv[src]*dinv[dst]
template <int LOGF>
__global__ void k_scatter(const float* __restrict__ h, const int* __restrict__ src,
                          const int* __restrict__ dst, const float* __restrict__ dinv,
                          float* __restrict__ agg, int E) {
    constexpr int F = 1 << LOGF;
    int tid = blockIdx.x * blockDim.x + threadIdx.x;
    int e = tid >> LOGF;
    int f = tid & (F - 1);
    if (e >= E) return;
    int s = src[e], d = dst[e];
    float norm = dinv[s] * dinv[d];
    atomicAdd(&agg[d * F + f], h[s * F + f] * norm);
}

template <int LOGF, bool RELU>
__global__ void k_bias_act(float* __restrict__ agg, const float* __restrict__ b, int n) {
    constexpr int F = 1 << LOGF;
    int i = blockIdx.x * blockDim.x + threadIdx.x;
    if (i >= n * F) return;
    float v = agg[i] + b[i & (F - 1)];
    if (RELU) v = fmaxf(v, 0.0f);
    agg[i] = v;
}

// ---- edge decoder: sigmoid(dot32(h[src], h[dst])) --------------------------

__global__ void k_decoder(const float* __restrict__ h, const int* __restrict__ src,
                          const int* __restrict__ dst, float* __restrict__ out, int E) {
    int e = blockIdx.x * blockDim.x + threadIdx.x;
    if (e >= E) return;
    const float4* hs = reinterpret_cast<const float4*>(h + (size_t)src[e] * 32);
    const float4* hd = reinterpret_cast<const float4*>(h + (size_t)dst[e] * 32);
    float acc = 0.0f;
#pragma unroll
    for (int i = 0; i < 8; ++i) {
        float4 a = hs[i], b = hd[i];
        acc += a.x * b.x + a.y * b.y + a.z * b.z + a.w * b.w;
    }
    out[e] = 1.0f / (1.0f + expf(-acc));
}

// ---------------------------------------------------------------------------

static inline unsigned blocks_for(long n) { return (unsigned)((n + TPB - 1) / TPB); }

extern "C" void kernel_launch(void* const* d_in, const int* in_sizes, int n_in,
                              void* d_out, int out_size, void* d_ws, size_t ws_size,
                              hipStream_t stream) {
    const float* x  = (const float*)d_in[0];
    const int*   ei = (const int*)d_in[1];
    const float* W1 = (const float*)d_in[2];
    const float* b1 = (const float*)d_in[3];
    const float* W2 = (const float*)d_in[4];
    const float* b2 = (const float*)d_in[5];
    const float* W3 = (const float*)d_in[6];
    const float* b3 = (const float*)d_in[7];

    const int Nn = in_sizes[0];       // 100000
    const int Ee = in_sizes[1] / 2;   // 1600000
    const int* src = ei;
    const int* dst = ei + Ee;
    float* out = (float*)d_out;

    // Workspace layout (floats): dinv[N] | A[32N] | B[32N] | C[32N]  (~38.8 MB)
    float* dinv = (float*)d_ws;
    float* A = dinv + Nn;
    float* B = A + (size_t)Nn * 32;
    float* C = B + (size_t)Nn * 32;

    // degrees -> dinv (deg init 1.0 accounts for self-loop)
    k_fill<<<blocks_for(Nn), TPB, 0, stream>>>(dinv, 1.0f, Nn);
    k_degree<<<blocks_for(Ee), TPB, 0, stream>>>(dst, dinv, Ee);
    k_rsqrt_inplace<<<blocks_for(Nn), TPB, 0, stream>>>(dinv, Nn);

    // ---- layer 1: 1 -> 8 (tmp=A, agg=B) ----
    k_lin1<<<blocks_for(Nn), TPB, 0, stream>>>(x, W1, A, Nn);
    k_selfloop_init<3><<<blocks_for((long)Nn * 8), TPB, 0, stream>>>(A, dinv, B, Nn);
    k_scatter<3><<<blocks_for((long)Ee * 8), TPB, 0, stream>>>(A, src, dst, dinv, B, Ee);
    k_bias_act<3, true><<<blocks_for((long)Nn * 8), TPB, 0, stream>>>(B, b1, Nn);

    // ---- layer 2: 8 -> 16 (in=B, tmp=C, agg=A) ----
    {
        int ntiles = (Nn + 15) >> 4;
        k_gemm_wmma<8, 16><<<blocks_for((long)ntiles * 32), TPB, 0, stream>>>(B, W2, C, Nn);
    }
    k_selfloop_init<4><<<blocks_for((long)Nn * 16), TPB, 0, stream>>>(C, dinv, A, Nn);
    k_scatter<4><<<blocks_for((long)Ee * 16), TPB, 0, stream>>>(C, src, dst, dinv, A, Ee);
    k_bias_act<4, true><<<blocks_for((long)Nn * 16), TPB, 0, stream>>>(A, b2, Nn);

    // ---- layer 3: 16 -> 32 (in=A, tmp=B, agg=C) ----
    {
        int ntiles = (Nn + 15) >> 4;
        k_gemm_wmma<16, 32><<<blocks_for((long)ntiles * 32), TPB, 0, stream>>>(A, W3, B, Nn);
    }
    k_selfloop_init<5><<<blocks_for((long)Nn * 32), TPB, 0, stream>>>(B, dinv, C, Nn);
    k_scatter<5><<<blocks_for((long)Ee * 32), TPB, 0, stream>>>(B, src, dst, dinv, C, Ee);
    k_bias_act<5, false><<<blocks_for((long)Nn * 32), TPB, 0, stream>>>(C, b3, Nn);

    // ---- edge decoder ----
    k_decoder<<<blocks_for(Ee), TPB, 0, stream>>>(C, src, dst, out, Ee);
}